// Corr_upsample_resize_norm_29609504538960
// MI455X (gfx1250) — compile-verified
//
#include <hip/hip_runtime.h>
#include <hip/hip_bf16.h>
#include <stdint.h>

// ---------------------------------------------------------------------------
// CDNA5 (gfx1250) pyramid correlation + leaky + downsample + global-max
// normalize. Heavy dot products run on V_WMMA_F32_16X16X4_F32 as banded
// Gram-matrix GEMMs over the channel dimension. All shape parameters are
// compile-time template constants so the K-loop uses immediate load offsets
// (no per-load 64-bit address adds) and stays predicate-free.
// ---------------------------------------------------------------------------

typedef float v2f __attribute__((ext_vector_type(2)));
typedef float v8f __attribute__((ext_vector_type(8)));

__device__ __forceinline__ unsigned encKey(float f) {
  unsigned u = __float_as_uint(f);
  return (u & 0x80000000u) ? ~u : (u | 0x80000000u);
}
__device__ __forceinline__ float decKey(unsigned k) {
  return (k & 0x80000000u) ? __uint_as_float(k & 0x7fffffffu)
                           : __uint_as_float(~k);
}

// Zero the used workspace region (max-key slots + cc volumes).
__global__ __launch_bounds__(256)
void zero_ws_kernel(float* __restrict__ p, int n) {
  int i = blockIdx.x * 256 + threadIdx.x;
  if (i < n) p[i] = 0.0f;
}

// One wave (32 lanes) per (b, y, dY, M-tile). Computes the 16x32 banded tile
// of G[x1,x2] = sum_c f1[c,y,x1] * f2[c,y+dY,x2] with K=C via WMMA f32
// 16x16x4, then scatters the in-bounds |x2-x1|<=R band into
// cc[b][dYi][dXi][y][x1]. OOB band slots were pre-zeroed.
template <int C, int W, int R>
__global__ __launch_bounds__(32)
void corr_coarse_wmma_kernel(const float* __restrict__ f1,
                             const float* __restrict__ f2,
                             float* __restrict__ cc)
{
  constexpr int nD = 2 * R + 1;
  constexpr int nT = (W + 15) / 16;
  constexpr int WW = W * W;

  const int lane = threadIdx.x & 31;
  const int m    = lane & 15;   // A-row / B-col / D-col index
  const int half = lane >> 4;   // lane-half selects K (A/B) or M+8 (D)

  int idx = blockIdx.x;
  const int t   = idx % nT;  idx /= nT;
  const int dYi = idx % nD;  idx /= nD;
  const int y   = idx % W;   idx /= W;
  const int b   = idx;                      // 0..1

  const int dY    = dYi - R;
  const int y2    = y + dY;
  const bool rowOK = (y2 >= 0) && (y2 < W);

  // A tile rows x1 in [16t, 16t+16); B tiles cols x2 in [16t-8, 16t+24).
  const int x1  = (t << 4) + m;
  const int x2a = (t << 4) - 8 + m;
  const int x2b = x2a + 16;

  // Clamped (always-valid) addresses; garbage values are never stored.
  const int x1c  = min(x1, W - 1);
  const int x2ac = min(max(x2a, 0), W - 1);
  const int x2bc = min(x2b, W - 1);          // x2b >= 8 always
  const int y2c  = min(max(y2, 0), W - 1);

  const size_t chanBase = (size_t)(b * C + (half << 1)) * (size_t)WW;
  const float* pA  = f1 + chanBase + (size_t)y * W + x1c;
  const float* pB0 = f2 + chanBase + (size_t)y2c * W + x2ac;
  const float* pB1 = f2 + chanBase + (size_t)y2c * W + x2bc;

  v8f acc0 = {};
  v8f acc1 = {};

  // Predicate-free hot loop: 6 plain loads (immediate offsets) + 2 WMMA per
  // K-step; constexpr strides let the compiler fold offsets into the load
  // instructions and keep a single pointer per matrix.
#pragma unroll 8
  for (int k = 0; k < C; k += 4) {
    v2f a, bb0, bb1;
    a.x   = pA[0];         // A[m][2*half + 0]
    a.y   = pA[WW];        // A[m][2*half + 1]
    bb0.x = pB0[0];        // B[2*half + 0][m]
    bb0.y = pB0[WW];       // B[2*half + 1][m]
    bb1.x = pB1[0];
    bb1.y = pB1[WW];

    acc0 = __builtin_amdgcn_wmma_f32_16x16x4_f32(false, a, false, bb0,
                                                 (short)0, acc0, false, false);
    acc1 = __builtin_amdgcn_wmma_f32_16x16x4_f32(false, a, false, bb1,
                                                 (short)0, acc1, false, false);
    pA  += 4 * WW;
    pB0 += 4 * WW;
    pB1 += 4 * WW;
  }

  if (!rowOK) return;   // whole tile row is zero-padding; cc pre-zeroed

  // D layout: VGPR v, lane: row M = v + 8*half, col N = lane&15.
  const size_t ccBase = ((size_t)(b * nD + dYi)) * (size_t)nD;
#pragma unroll
  for (int v = 0; v < 8; ++v) {
    const int M   = v + (half << 3);
    const int x1v = (t << 4) + M;
    if (x1v < W) {
      {                                      // acc0: x2 = 16t - 8 + m
        const int x2 = (t << 4) - 8 + m;
        const int dX = x2 - x1v;
        if (dX >= -R && dX <= R && x2 >= 0 && x2 < W)
          cc[(ccBase + (size_t)(dX + R)) * WW + (size_t)y * W + x1v] = acc0[v];
      }
      {                                      // acc1: x2 = 16t + 8 + m
        const int x2 = (t << 4) + 8 + m;
        const int dX = x2 - x1v;
        if (dX >= -R && dX <= R && x2 < W)
          cc[(ccBase + (size_t)(dX + R)) * WW + (size_t)y * W + x1v] = acc1[v];
      }
    }
  }
}

// One thread per fine corr element: gather coarse value, LeakyReLU, write
// corr; downsample points also write unnormalized dw and contribute to the
// per-level running max (order-preserving uint key, block-reduced).
template <int W, int L, int R>
__global__ __launch_bounds__(256)
void gather_out_kernel(const float* __restrict__ cc, float* __restrict__ out,
                       size_t corrOff, size_t dwOff,
                       unsigned* __restrict__ maxKey)
{
  constexpr int nD = 2 * R + 1;
  constexpr int FS = 96;
  constexpr int total = 2 * 81 * FS * FS;
  const int tid = blockIdx.x * blockDim.x + threadIdx.x;

  unsigned key = 0u;
  if (tid < total) {
    int x  = tid % FS;
    int t1 = tid / FS;
    int y  = t1 % FS;
    int t2 = t1 / FS;
    int d  = t2 % 81;
    int b  = t2 / 81;
    int di = d / 9, dj = d % 9;
    int ty = y + di - 4;
    int tx = x + dj - 4;
    int Y = y >> L, X = x >> L;
    int dY = (ty >> L) - Y;     // arithmetic shift == floor-div for pow2
    int dX = (tx >> L) - X;     // guaranteed within [-R, R]
    size_t ci = (((size_t)(b * nD + dY + R)) * (size_t)nD + (size_t)(dX + R))
                    * (size_t)(W * W)
                + (size_t)Y * W + X;
    float v   = cc[ci];
    float val = (v >= 0.0f) ? v : 0.01f * v;   // LeakyReLU(0.01)
    out[corrOff + (size_t)tid] = val;

    constexpr int mask = (1 << L) - 1;
    if (((y | x) & mask) == 0) {               // downsample point
      out[dwOff + (((size_t)(b * 81 + d)) * W + Y) * W + X] = val;
      key = encKey(val);
    }
  }

  __shared__ unsigned red[256];
  red[threadIdx.x] = key;
  __syncthreads();
#pragma unroll
  for (int s = 128; s > 0; s >>= 1) {
    if ((int)threadIdx.x < s) {
      unsigned o = red[threadIdx.x + s];
      if (o > red[threadIdx.x]) red[threadIdx.x] = o;
    }
    __syncthreads();
  }
  if (threadIdx.x == 0 && red[0] != 0u) atomicMax(maxKey, red[0]);
}

__global__ __launch_bounds__(256)
void scale_dw_kernel(float* __restrict__ out, size_t dwOff, int n,
                     const unsigned* __restrict__ key)
{
  const int tid = blockIdx.x * blockDim.x + threadIdx.x;
  if (tid < n) {
    float mx = decKey(*key);
    out[dwOff + (size_t)tid] = out[dwOff + (size_t)tid] / mx;
  }
}

// ---------------------------------------------------------------------------
// Host-side dispatch
// ---------------------------------------------------------------------------

template <int C, int W, int R>
static void launch_corr(const float* f1, const float* f2, float* cc,
                        hipStream_t stream) {
  constexpr int nD = 2 * R + 1;
  constexpr int nT = (W + 15) / 16;
  const int grid = nT * nD * W * 2;   // (tile, dY, y, b)
  corr_coarse_wmma_kernel<C, W, R><<<grid, 32, 0, stream>>>(f1, f2, cc);
}

template <int W, int L, int R>
static void launch_gather(const float* cc, float* out, size_t corrOff,
                          size_t dwOff, unsigned* key, hipStream_t stream) {
  constexpr int total = 2 * 81 * 96 * 96;
  gather_out_kernel<W, L, R><<<(total + 255) / 256, 256, 0, stream>>>(
      cc, out, corrOff, dwOff, key);
}

extern "C" void kernel_launch(void* const* d_in, const int* in_sizes, int n_in,
                              void* d_out, int out_size, void* d_ws,
                              size_t ws_size, hipStream_t stream)
{
  (void)in_sizes; (void)n_in; (void)out_size; (void)ws_size;
  static const int WL[6] = {96, 48, 24, 12, 6, 3};
  static const int RL[6] = {4, 2, 1, 1, 1, 1};  // coarse displacement radius

  unsigned* keys  = (unsigned*)d_ws;            // 6 max-key slots (first 64B)
  float* ccbase   = (float*)((char*)d_ws + 64); // coarse corr volumes (~6.5 MB)
  float* out      = (float*)d_out;

  size_t cco[6], dwo[6], co[6];
  size_t ccAcc = 0, oAcc = 0;
  for (int i = 0; i < 6; ++i) {
    const int W = WL[i], nD = 2 * RL[i] + 1;
    cco[i] = ccAcc;  ccAcc += (size_t)2 * nD * nD * W * W;
    dwo[i] = oAcc;   oAcc  += (size_t)2 * 81 * W * W;        // dw_i
    co[i]  = oAcc;   oAcc  += (size_t)2 * 81 * 96 * 96;      // corr_i
  }

  // Zero keys + cc volumes (16 floats of key space + cc region).
  const int zn = (int)(16 + ccAcc);
  zero_ws_kernel<<<(zn + 255) / 256, 256, 0, stream>>>((float*)d_ws, zn);

  // Pass 1: coarse correlation volumes via fp32 WMMA banded GEMMs.
  const float* f1[6];
  const float* f2[6];
  for (int i = 0; i < 6; ++i) {
    f1[i] = (const float*)d_in[2 * i];       // x_i
    f2[i] = (const float*)d_in[2 * i + 1];   // xp_i
  }
  launch_corr< 512, 96, 4>(f1[0], f2[0], ccbase + cco[0], stream);
  launch_corr<1024, 48, 2>(f1[1], f2[1], ccbase + cco[1], stream);
  launch_corr< 512, 24, 1>(f1[2], f2[2], ccbase + cco[2], stream);
  launch_corr< 256, 12, 1>(f1[3], f2[3], ccbase + cco[3], stream);
  launch_corr< 256,  6, 1>(f1[4], f2[4], ccbase + cco[4], stream);
  launch_corr< 256,  3, 1>(f1[5], f2[5], ccbase + cco[5], stream);

  // Pass 2: gather -> leaky -> corr + unnormalized dw + per-level max.
  launch_gather<96, 0, 4>(ccbase + cco[0], out, co[0], dwo[0], keys + 0, stream);
  launch_gather<48, 1, 2>(ccbase + cco[1], out, co[1], dwo[1], keys + 1, stream);
  launch_gather<24, 2, 1>(ccbase + cco[2], out, co[2], dwo[2], keys + 2, stream);
  launch_gather<12, 3, 1>(ccbase + cco[3], out, co[3], dwo[3], keys + 3, stream);
  launch_gather< 6, 4, 1>(ccbase + cco[4], out, co[4], dwo[4], keys + 4, stream);
  launch_gather< 3, 5, 1>(ccbase + cco[5], out, co[5], dwo[5], keys + 5, stream);

  // Pass 3: normalize dw by global max.
  for (int i = 0; i < 6; ++i) {
    const int W = WL[i];
    const int n = 2 * 81 * W * W;
    scale_dw_kernel<<<(n + 255) / 256, 256, 0, stream>>>(out, dwo[i], n,
                                                         keys + i);
  }
}